// LearnedQueryAttention_70102456205325
// MI455X (gfx1250) — compile-verified
//
#include <hip/hip_runtime.h>
#include <hip/hip_bf16.h>
#include <math.h>

typedef __attribute__((ext_vector_type(2))) float v2f;
typedef __attribute__((ext_vector_type(8))) float v8f;

#define B_   4
#define S_   4096
#define D_   512
#define H_   8
#define HD_  64
#define SEG_ 256
#define L_   256   // SEG_ * Q, Q == 1

#define WMMA4(a, b, c) \
  __builtin_amdgcn_wmma_f32_16x16x4_f32(false, (a), false, (b), (short)0, (c), false, false)

// -----------------------------------------------------------------------------
// C[M x 512] = A[M x 512] @ W^T, W row-major [512 x 512] (torch Linear weight).
// One wave computes a 16(M) x 64(N) strip: 4 WMMA accumulators sharing one
// A fragment per K-step -> 1.25 loads per v_wmma_f32_16x16x4_f32.
// A frag (16x4): lane<16 row=lane K={k,k+1}; lane>=16 K={k+2,k+3}.
// B frag (4x16): lane col = lane&15, same K-pair split. B[k][n] = W[n][k].
// C/D (16x16):   VGPR j holds row half*8+j, col = lane&15.
// -----------------------------------------------------------------------------
__global__ __launch_bounds__(256) void gemm_xt_wmma(const float* __restrict__ A,
                                                    const float* __restrict__ W,
                                                    float* __restrict__ C, int M) {
  const int lane  = threadIdx.x & 31;
  const int wave  = blockIdx.x * 8 + (threadIdx.x >> 5);
  const int strip = wave & 7;                // D_/64 = 8 N-strips
  const int tileM = wave >> 3;
  if (tileM * 16 >= M) return;               // wave-uniform

  const int r    = lane & 15;
  const int half = lane >> 4;

  const float* arow = A + (size_t)(tileM * 16 + r) * D_ + 2 * half;
  const float* w0   = W + (size_t)(strip * 64 +  0 + r) * D_ + 2 * half;
  const float* w1   = w0 + (size_t)16 * D_;
  const float* w2   = w0 + (size_t)32 * D_;
  const float* w3   = w0 + (size_t)48 * D_;

  v8f acc0 = {0.f,0.f,0.f,0.f,0.f,0.f,0.f,0.f};
  v8f acc1 = acc0, acc2 = acc0, acc3 = acc0;

#pragma unroll 4
  for (int k = 0; k < D_; k += 4) {
    v2f a  = *(const v2f*)(arow + k);
    v2f b0 = *(const v2f*)(w0 + k);
    v2f b1 = *(const v2f*)(w1 + k);
    v2f b2 = *(const v2f*)(w2 + k);
    v2f b3 = *(const v2f*)(w3 + k);
    acc0 = WMMA4(a, b0, acc0);
    acc1 = WMMA4(a, b1, acc1);
    acc2 = WMMA4(a, b2, acc2);
    acc3 = WMMA4(a, b3, acc3);
  }

  float* cbase = C + (size_t)(tileM * 16 + half * 8) * D_ + strip * 64 + r;
#pragma unroll
  for (int j = 0; j < 8; ++j) {
    float* cj = cbase + (size_t)j * D_;
    cj[ 0] = acc0[j];
    cj[16] = acc1[j];
    cj[32] = acc2[j];
    cj[48] = acc3[j];
  }
}

// q_vec[n] = (sum_d qb[d] * w_q[n,d]) * (1/sqrt(hd));  n in [0,512)
__global__ __launch_bounds__(256) void qproj_kernel(const float* __restrict__ qb,
                                                    const float* __restrict__ wq,
                                                    float* __restrict__ QV) {
  int n = blockIdx.x * 256 + threadIdx.x;
  if (n >= D_) return;
  const float4* w = (const float4*)(wq + (size_t)n * D_);
  const float4* q = (const float4*)qb;
  float acc = 0.f;
#pragma unroll 4
  for (int i = 0; i < D_ / 4; ++i) {
    float4 a = w[i], b = q[i];
    acc += a.x * b.x + a.y * b.y + a.z * b.z + a.w * b.w;
  }
  QV[n] = acc * 0.125f;  // 1/sqrt(64)
}

// Fold query through w_k: qk[h,d] = sum_{j<64} QV[h*64+j] * w_k[h*64+j, d]
// (rank-1 query => K never needs to be materialized)
__global__ __launch_bounds__(256) void qk_kernel(const float* __restrict__ QV,
                                                 const float* __restrict__ wk,
                                                 float* __restrict__ QK) {
  int i = blockIdx.x * 256 + threadIdx.x;   // < H*D = 4096
  int h = i >> 9;
  int d = i & 511;
  const float* qv = QV + h * HD_;
  const float* wb = wk + (size_t)(h * HD_) * D_ + d;
  float acc = 0.f;
#pragma unroll 8
  for (int j = 0; j < HD_; ++j) acc += qv[j] * wb[(size_t)j * D_];
  QK[i] = acc;
}

// score[b,h,s] = x[b,s,:] . qk[h,:]   (scale folded into QV -> QK)
__global__ __launch_bounds__(256) void scores_kernel(const float* __restrict__ x,
                                                     const float* __restrict__ QK,
                                                     float* __restrict__ E) {
  int idx = blockIdx.x * 256 + threadIdx.x;     // < B*H*S = 131072
  int b   = idx >> 15;                          // H*S = 32768
  int rem = idx & 32767;
  int h   = rem >> 12;                          // S = 4096
  int s   = rem & 4095;
  const float4* xp = (const float4*)(x + ((size_t)(b * S_ + s)) * D_);
  const float4* qp = (const float4*)(QK + h * D_);
  float acc = 0.f;
#pragma unroll 8
  for (int i = 0; i < D_ / 4; ++i) {
    float4 xv = xp[i], qv = qp[i];
    acc += xv.x * qv.x + xv.y * qv.y + xv.z * qv.z + xv.w * qv.w;
  }
  E[((size_t)(b * H_ + h)) * S_ + s] = acc;
}

// seg_id sorted per row -> start offsets of each segment: ST[b][l], ST[b][SEG]=S
__global__ __launch_bounds__(256) void bounds_kernel(const int* __restrict__ seg_id,
                                                     int* __restrict__ ST) {
  int idx = blockIdx.x * 256 + threadIdx.x;     // < B*S
  int b = idx >> 12;
  int s = idx & 4095;
  int seg  = seg_id[b * S_ + s];
  int prev = (s == 0) ? -1 : seg_id[b * S_ + s - 1];
  for (int l = prev + 1; l <= seg; ++l) ST[b * (SEG_ + 1) + l] = s;
  if (s == S_ - 1)
    for (int l = seg + 1; l <= SEG_; ++l) ST[b * (SEG_ + 1) + l] = S_;
}

// Per (b,h): row max, e = exp(score - max) (in place), T_den = sum e,
// T_num[d] = sum_s e[s] * V[b,s,h*64+d]
__global__ __launch_bounds__(256) void rowstats_kernel(float* __restrict__ E,
                                                       const float* __restrict__ V,
                                                       float* __restrict__ TN,
                                                       float* __restrict__ TD) {
  __shared__ float sm[256];
  const int bh = blockIdx.x;            // 0..31
  const int b  = bh >> 3;
  const int h  = bh & 7;
  const int tid = threadIdx.x;
  float* sc = E + (size_t)bh * S_;

  // pass 1: max
  float m = -3.4e38f;
  for (int s = tid; s < S_; s += 256) m = fmaxf(m, sc[s]);
  sm[tid] = m; __syncthreads();
  for (int off = 128; off > 0; off >>= 1) {
    if (tid < off) sm[tid] = fmaxf(sm[tid], sm[tid + off]);
    __syncthreads();
  }
  const float mmax = sm[0];
  __syncthreads();

  // pass 2: exp in place + denominator
  float den = 0.f;
  for (int s = tid; s < S_; s += 256) {
    float ev = __expf(sc[s] - mmax);
    sc[s] = ev;
    den += ev;
  }
  sm[tid] = den; __syncthreads();
  for (int off = 128; off > 0; off >>= 1) {
    if (tid < off) sm[tid] += sm[tid + off];
    __syncthreads();
  }
  if (tid == 0) TD[bh] = sm[0];
  __syncthreads();

  // pass 3: T_num[d] = sum_s e[s] * V[b,s,h*64+d]
  const int d = tid & 63;
  const int g = tid >> 6;               // 4 groups
  float acc = 0.f;
  const float* vp = V + (size_t)(b * S_) * D_ + h * HD_ + d;
  for (int s = g; s < S_; s += 4) acc += sc[s] * vp[(size_t)s * D_];
  sm[g * 64 + d] = acc; __syncthreads();
  if (g == 0) TN[bh * HD_ + d] = sm[d] + sm[64 + d] + sm[128 + d] + sm[192 + d];
}

// out_heads[b,l,h*64+d] = (T_num - seg_sum(e*v)) / (T_den - seg_sum(e))
__global__ __launch_bounds__(256) void outheads_kernel(const float* __restrict__ E,
                                                       const float* __restrict__ V,
                                                       const float* __restrict__ TN,
                                                       const float* __restrict__ TD,
                                                       const int* __restrict__ ST,
                                                       const unsigned char* __restrict__ valid,
                                                       float* __restrict__ OH) {
  const int b = blockIdx.x >> 8;        // L_ = 256
  const int l = blockIdx.x & 255;
  const int s0 = ST[b * (SEG_ + 1) + l];
  const int s1 = ST[b * (SEG_ + 1) + l + 1];
  const unsigned char* vm = valid + (size_t)b * S_;
  for (int i = threadIdx.x; i < D_; i += 256) {
    const int h = i >> 6, d = i & 63;
    const int bh = b * H_ + h;
    float num = TN[bh * HD_ + d];
    float den = TD[bh];
    const float* ev = E + (size_t)bh * S_;
    const float* vp = V + (size_t)(b * S_) * D_ + h * HD_ + d;
    for (int s = s0; s < s1; ++s) {
      if (vm[s]) {                      // only valid keys in segment l are masked out
        float e = ev[s];
        num -= e * vp[(size_t)s * D_];
        den -= e;
      }
    }
    OH[((size_t)(b * L_ + l)) * D_ + i] = num / den;
  }
}

extern "C" void kernel_launch(void* const* d_in, const int* in_sizes, int n_in,
                              void* d_out, int out_size, void* d_ws, size_t ws_size,
                              hipStream_t stream) {
  const float*         x     = (const float*)d_in[0];
  const int*           seg   = (const int*)d_in[1];
  const unsigned char* valid = (const unsigned char*)d_in[2];
  // d_in[3] = s_seg_max (256, hardcoded)
  const float*         qb    = (const float*)d_in[4];
  const float*         w_q   = (const float*)d_in[5];
  const float*         w_k   = (const float*)d_in[6];
  const float*         w_v   = (const float*)d_in[7];
  const float*         w_o   = (const float*)d_in[8];
  float*               out   = (float*)d_out;

  // workspace layout (floats)
  float* ws = (float*)d_ws;
  float* V  = ws;                          // B*S*D
  float* OH = V  + (size_t)B_ * S_ * D_;   // B*L*D
  float* E  = OH + (size_t)B_ * L_ * D_;   // B*H*S (scores, then exp in place)
  float* QV = E  + (size_t)B_ * H_ * S_;   // D
  float* QK = QV + D_;                     // H*D
  float* TN = QK + (size_t)H_ * D_;        // B*H*HD
  float* TD = TN + (size_t)B_ * H_ * HD_;  // B*H
  int*   ST = (int*)(TD + B_ * H_);        // B*(SEG+1)

  qproj_kernel<<<2, 256, 0, stream>>>(qb, w_q, QV);
  qk_kernel<<<(H_ * D_) / 256, 256, 0, stream>>>(QV, w_k, QK);

  // V projection: waves = (M/16)*(512/64) -> blocks = M/16 = 1024
  gemm_xt_wmma<<<(B_ * S_) / 16, 256, 0, stream>>>(x, w_v, V, B_ * S_);

  scores_kernel<<<(B_ * H_ * S_) / 256, 256, 0, stream>>>(x, QK, E);
  bounds_kernel<<<(B_ * S_) / 256, 256, 0, stream>>>(seg, ST);
  rowstats_kernel<<<B_ * H_, 256, 0, stream>>>(E, V, TN, TD);
  outheads_kernel<<<B_ * L_, 256, 0, stream>>>(E, V, TN, TD, ST, valid, OH);

  // final projection: M = B*L = 1024 -> 64 blocks
  gemm_xt_wmma<<<(B_ * L_) / 16, 256, 0, stream>>>(OH, w_o, out, B_ * L_);
}